// MessageAttention_34359738938
// MI455X (gfx1250) — compile-verified
//
#include <hip/hip_runtime.h>
#include <hip/hip_bf16.h>

// Problem constants (fixed by the reference).
#define NHEADS 4

typedef float v2f __attribute__((ext_vector_type(2)));
typedef float v8f __attribute__((ext_vector_type(8)));

// Force global (AS1) memory ops for pointers we know are global, so the
// backend emits global_load/global_store instead of flat_* (flat ops tick
// both LOADcnt and DScnt and share the LDS datapath on CDNA5).
typedef const float __attribute__((address_space(1))) * gcptr;
typedef float __attribute__((address_space(1))) * gptr;

__device__ __forceinline__ float gload(const float* p) {
  return *(gcptr)p;
}
__device__ __forceinline__ void gstore(float* p, float v) {
  *(gptr)p = v;
}

// ---------------------------------------------------------------------------
// WMMA helpers: V_WMMA_F32_16X16X4_F32, D = A(16x4) * B(4x16) + C(16x16)
// A frag (ISA 7.12.2, 32-bit A 16x4): lane 0-15 -> M=lane, VGPR0=K0,VGPR1=K1;
//                                     lane16-31 -> M=lane-16, VGPR0=K2,VGPR1=K3.
// B frag (4x16, row striped across lanes): lane = N, half selects K pair.
// C/D frag: VGPR r holds M=r (lanes 0-15) / M=r+8 (lanes 16-31), N = lane&15.
// ---------------------------------------------------------------------------
__device__ __forceinline__ v8f wmma_f32x4(v2f a, v2f b, v8f c) {
  return __builtin_amdgcn_wmma_f32_16x16x4_f32(
      /*neg_a=*/false, a, /*neg_b=*/false, b,
      /*c_mod=*/(short)0, c, /*reuse_a=*/false, /*reuse_b=*/false);
}

// One 16x16 output block of C = A(16xK, LDS, row-major lda) * W^T where the
// weight W is row-major [Nout][K] in global memory (so B[k][n] = W[n][k]).
__device__ __forceinline__ v8f gemm_block(const float* __restrict__ A, int lda,
                                          int K, const float* __restrict__ W,
                                          int ldw, int nb, int lane) {
  const int rc = lane & 15;          // M for A, N for B
  const int kb = (lane >> 4) << 1;   // K sub-pair select: 0 or 2
  v8f acc = {0.f, 0.f, 0.f, 0.f, 0.f, 0.f, 0.f, 0.f};
#pragma unroll
  for (int k0 = 0; k0 < K; k0 += 4) {
    v2f a, b;
    a.x = A[rc * lda + k0 + kb];
    a.y = A[rc * lda + k0 + kb + 1];
    b.x = gload(&W[(nb + rc) * ldw + k0 + kb]);
    b.y = gload(&W[(nb + rc) * ldw + k0 + kb + 1]);
    acc = wmma_f32x4(a, b, acc);
  }
  return acc;
}

__device__ __forceinline__ void store_block(v8f acc, float* C, int ldc,
                                            int col0, int lane) {
  const int rc = lane & 15;
  const int rbase = (lane < 16) ? 0 : 8;
#pragma unroll
  for (int r = 0; r < 8; ++r)
    C[(rbase + r) * ldc + col0 + rc] = acc[r];
}

// ---------------------------------------------------------------------------
// GDBLinear for a 16-node tile, one wave32. All matmuls via WMMA f32 16x16x4.
// ---------------------------------------------------------------------------
__device__ __forceinline__ void gdb_eval(
    int n0, int lane,
    const float avec[3][16][32],  // [t][m][c]
    const float asca[16][64],
    float vinter[3][16][32], float cc[16][64],
    float os[16][64], float gate[16][32],
    const float* __restrict__ Wv1,   // [32][32]
    const float* __restrict__ Wv2,   // [32][32]
    const float* __restrict__ Ws1,   // [32][64]
    const float* __restrict__ Ws2,   // [64][64]
    const float* __restrict__ Wg,    // [32][64]
    const float* __restrict__ bg,    // [32]
    float* __restrict__ out_s,       // [N][64]
    float* __restrict__ out_v) {     // [N][32][3]
  // 1) vinter[t] = avec[t] @ Wv1^T   (16x32 @ 32x32)
  for (int t = 0; t < 3; ++t)
    for (int nb = 0; nb < 32; nb += 16) {
      v8f acc = gemm_block(&avec[t][0][0], 32, 32, Wv1, 32, nb, lane);
      store_block(acc, &vinter[t][0][0], 32, nb, lane);
    }
  // 2) v_norm -> cc[:, 0:32]
  for (int i = lane; i < 16 * 32; i += 32) {
    int m = i >> 5, h = i & 31;
    float x = vinter[0][m][h], y = vinter[1][m][h], z = vinter[2][m][h];
    cc[m][h] = sqrtf(x * x + y * y + z * z);
  }
  // 3) z_sca = asca @ Ws1^T -> cc[:, 32:64]   (16x64 @ 64x32)
  for (int nb = 0; nb < 32; nb += 16) {
    v8f acc = gemm_block(&asca[0][0], 64, 64, Ws1, 64, nb, lane);
    store_block(acc, &cc[0][0], 64, 32 + nb, lane);
  }
  // 4) out_sca = cc @ Ws2^T -> os (and global)   (16x64 @ 64x64)
  for (int nb = 0; nb < 64; nb += 16) {
    v8f acc = gemm_block(&cc[0][0], 64, 64, Ws2, 64, nb, lane);
    store_block(acc, &os[0][0], 64, nb, lane);
    const int rc = lane & 15;
    const int rbase = (lane < 16) ? 0 : 8;
#pragma unroll
    for (int r = 0; r < 8; ++r)
      gstore(&out_s[(size_t)(n0 + rbase + r) * 64 + nb + rc], acc[r]);
  }
  // 5) gate = sigmoid(os @ Wg^T + bg)   (16x64 @ 64x32)
  for (int nb = 0; nb < 32; nb += 16) {
    v8f acc = gemm_block(&os[0][0], 64, 64, Wg, 64, nb, lane);
    store_block(acc, &gate[0][0], 32, nb, lane);
  }
  for (int i = lane; i < 16 * 32; i += 32) {
    int m = i >> 5, g = i & 31;
    float v = gate[m][g] + gload(&bg[g]);
    gate[m][g] = 1.0f / (1.0f + __expf(-v));
  }
  // 6) out_vec[m][o][t] = gate[m][o] * (vinter[t] @ Wv2^T)[m][o]
  for (int t = 0; t < 3; ++t)
    for (int nb = 0; nb < 32; nb += 16) {
      v8f acc = gemm_block(&vinter[t][0][0], 32, 32, Wv2, 32, nb, lane);
      const int rc = lane & 15;
      const int rbase = (lane < 16) ? 0 : 8;
#pragma unroll
      for (int r = 0; r < 8; ++r) {
        int m = rbase + r;
        int o = nb + rc;
        gstore(&out_v[(size_t)(n0 + m) * 96 + o * 3 + t], gate[m][o] * acc[r]);
      }
    }
}

// One wave32 per 16-node tile; computes K path (to ws) and root/V path (to out).
__global__ void __launch_bounds__(32)
node_kernel(const float* __restrict__ x_sca, const float* __restrict__ x_vec,
            const float* kWv1, const float* kWv2, const float* kWs1,
            const float* kWs2, const float* kWg, const float* kbg,
            const float* vWv1, const float* vWv2, const float* vWs1,
            const float* vWs2, const float* vWg, const float* vbg,
            float* __restrict__ ks, float* __restrict__ kv,
            float* __restrict__ out, int nnodes) {
  __shared__ float avec[3][16][32];
  __shared__ float asca[16][64];
  __shared__ float vinter[3][16][32];
  __shared__ float cc[16][64];
  __shared__ float os[16][64];
  __shared__ float gate[16][32];
  const int lane = threadIdx.x;
  const int n0 = blockIdx.x * 16;

  // Stage the 16-node tile (x_vec is [n][c][t] -> avec[t][m][c]).
  for (int i = lane; i < 16 * 96; i += 32) {
    int m = i / 96, r = i % 96, c = r / 3, t = r % 3;
    avec[t][m][c] = gload(&x_vec[(size_t)(n0 + m) * 96 + r]);
  }
  for (int i = lane; i < 16 * 64; i += 32) {
    int m = i >> 6, j = i & 63;
    asca[m][j] = gload(&x_sca[(size_t)(n0 + m) * 64 + j]);
  }

  // K path: features gathered by edges later.
  gdb_eval(n0, lane, avec, asca, vinter, cc, os, gate,
           kWv1, kWv2, kWs1, kWs2, kWg, kbg, ks, kv);
  // V/root path: written straight into d_out; edge messages atomicAdd on top.
  gdb_eval(n0, lane, avec, asca, vinter, cc, os, gate,
           vWv1, vWv2, vWs1, vWs2, vWg, vbg, out, out + (size_t)nnodes * 64);
}

// ---------------------------------------------------------------------------
// Edge-side scatter-softmax (bandwidth/atomic bound).
// ---------------------------------------------------------------------------
__device__ __forceinline__ void atomicMaxF(float* addr, float val) {
  int* ai = (int*)addr;
  int old = __float_as_int(*addr);
  while (__int_as_float(old) < val) {
    int assumed = old;
    old = atomicCAS(ai, assumed, __float_as_int(val));
    if (old == assumed) break;
  }
}

__global__ void init_kernel(float* max_s, float* max_v, float* sum_s,
                            float* sum_v, int n) {
  int i = blockIdx.x * blockDim.x + threadIdx.x;
  if (i < n) {
    float ninf = __int_as_float(0xff800000u);
    max_s[i] = ninf;
    max_v[i] = ninf;
    sum_s[i] = 0.f;
    sum_v[i] = 0.f;
  }
}

// Thread per (edge, head): raw attention logits + per-node running max.
__global__ void edge_alpha_kernel(const float* __restrict__ q_sca,
                                  const float* __restrict__ q_vec,
                                  const int* __restrict__ idx,
                                  const float* __restrict__ ks,
                                  const float* __restrict__ kv,
                                  float* __restrict__ alpha_s,
                                  float* __restrict__ alpha_v,
                                  float* max_s, float* max_v, int nedges) {
  int tid = blockIdx.x * blockDim.x + threadIdx.x;
  if (tid >= nedges * NHEADS) return;
  int e = tid >> 2, h = tid & 3;
  int i = idx[e];
  const float* qs = q_sca + (size_t)e * 64 + h * 16;
  const float* xs = ks + (size_t)i * 64 + h * 16;
  const float* qv = q_vec + (size_t)e * 96 + h * 24;
  const float* xv = kv + (size_t)i * 96 + h * 24;
  __builtin_prefetch(qv, 0, 1);
  __builtin_prefetch(xv, 0, 1);
  float as = 0.f;
#pragma unroll
  for (int j = 0; j < 16; ++j) as += qs[j] * xs[j];
  as *= 0.25f;  // 1/sqrt(16)
  float av = 0.f;
#pragma unroll
  for (int j = 0; j < 24; ++j) av += qv[j] * xv[j];
  av *= 0.35355339059327373f;  // 1/sqrt(8)
  alpha_s[tid] = as;
  alpha_v[tid] = av;
  atomicMaxF(&max_s[i * 4 + h], as);
  atomicMaxF(&max_v[i * 4 + h], av);
}

// Thread per (edge, head): exp(alpha - max), per-node sums.
__global__ void edge_exp_kernel(const int* __restrict__ idx,
                                float* __restrict__ alpha_s,
                                float* __restrict__ alpha_v,
                                const float* __restrict__ max_s,
                                const float* __restrict__ max_v,
                                float* sum_s, float* sum_v, int nedges) {
  int tid = blockIdx.x * blockDim.x + threadIdx.x;
  if (tid >= nedges * NHEADS) return;
  int e = tid >> 2, h = tid & 3;
  int i = idx[e];
  float es = __expf(alpha_s[tid] - max_s[i * 4 + h]);
  float ev = __expf(alpha_v[tid] - max_v[i * 4 + h]);
  alpha_s[tid] = es;  // now holds exp term
  alpha_v[tid] = ev;
  atomicAdd(&sum_s[i * 4 + h], es);
  atomicAdd(&sum_v[i * 4 + h], ev);
}

// Warp per edge: normalize and scatter weighted message onto out (root already there).
__global__ void __launch_bounds__(256)
edge_scatter_kernel(const float* __restrict__ q_sca,
                    const float* __restrict__ q_vec,
                    const int* __restrict__ idx,
                    const float* __restrict__ alpha_s,
                    const float* __restrict__ alpha_v,
                    const float* __restrict__ sum_s,
                    const float* __restrict__ sum_v,
                    float* __restrict__ out, int nedges, int nnodes) {
  int e = blockIdx.x * 8 + (threadIdx.x >> 5);
  int lane = threadIdx.x & 31;
  if (e >= nedges) return;
  int i = idx[e];
  float* out_v = out + (size_t)nnodes * 64;
#pragma unroll
  for (int rep = 0; rep < 2; ++rep) {
    int j = lane + rep * 32;
    int h = j >> 4;
    float w = alpha_s[e * 4 + h] / sum_s[i * 4 + h];
    atomicAdd(&out[(size_t)i * 64 + j], w * q_sca[(size_t)e * 64 + j]);
  }
#pragma unroll
  for (int rep = 0; rep < 3; ++rep) {
    int j = lane + rep * 32;
    int h = j / 24;  // 8 vec channels * 3 comps per head
    float w = alpha_v[e * 4 + h] / sum_v[i * 4 + h];
    atomicAdd(&out_v[(size_t)i * 96 + j], w * q_vec[(size_t)e * 96 + j]);
  }
}

// ---------------------------------------------------------------------------
extern "C" void kernel_launch(void* const* d_in, const int* in_sizes, int n_in,
                              void* d_out, int out_size, void* d_ws,
                              size_t ws_size, hipStream_t stream) {
  const float* x_sca = (const float*)d_in[0];
  const float* x_vec = (const float*)d_in[1];
  const float* q_sca = (const float*)d_in[2];
  const float* q_vec = (const float*)d_in[3];
  const int* eidx = (const int*)d_in[4];
  const float* kWv1 = (const float*)d_in[5];
  const float* kWv2 = (const float*)d_in[6];
  const float* kWs1 = (const float*)d_in[7];
  const float* kWs2 = (const float*)d_in[8];
  const float* kWg = (const float*)d_in[9];
  const float* kbg = (const float*)d_in[10];
  const float* vWv1 = (const float*)d_in[11];
  const float* vWv2 = (const float*)d_in[12];
  const float* vWs1 = (const float*)d_in[13];
  const float* vWs2 = (const float*)d_in[14];
  const float* vWg = (const float*)d_in[15];
  const float* vbg = (const float*)d_in[16];

  const int nnodes = in_sizes[0] / 64;  // 50000 (divisible by 16)
  const int nedges = in_sizes[4];       // 800000

  float* ws = (float*)d_ws;
  float* ks = ws;                                   // [N][64]
  float* kv = ks + (size_t)nnodes * 64;             // [N][32][3]
  float* alpha_s = kv + (size_t)nnodes * 96;        // [E][4]
  float* alpha_v = alpha_s + (size_t)nedges * 4;    // [E][4]
  float* max_s = alpha_v + (size_t)nedges * 4;      // [N][4]
  float* max_v = max_s + (size_t)nnodes * 4;
  float* sum_s = max_v + (size_t)nnodes * 4;
  float* sum_v = sum_s + (size_t)nnodes * 4;
  float* out = (float*)d_out;

  const int nh = nnodes * NHEADS;
  init_kernel<<<(nh + 255) / 256, 256, 0, stream>>>(max_s, max_v, sum_s, sum_v,
                                                    nh);
  node_kernel<<<nnodes / 16, 32, 0, stream>>>(
      x_sca, x_vec, kWv1, kWv2, kWs1, kWs2, kWg, kbg, vWv1, vWv2, vWs1, vWs2,
      vWg, vbg, ks, kv, out, nnodes);
  const int eh = nedges * NHEADS;
  edge_alpha_kernel<<<(eh + 255) / 256, 256, 0, stream>>>(
      q_sca, q_vec, eidx, ks, kv, alpha_s, alpha_v, max_s, max_v, nedges);
  edge_exp_kernel<<<(eh + 255) / 256, 256, 0, stream>>>(
      eidx, alpha_s, alpha_v, max_s, max_v, sum_s, sum_v, nedges);
  edge_scatter_kernel<<<(nedges + 7) / 8, 256, 0, stream>>>(
      q_sca, q_vec, eidx, alpha_s, alpha_v, sum_s, sum_v, out, nedges, nnodes);
}